// RingBufferKVCache_75471165325702
// MI455X (gfx1250) — compile-verified
//
#include <hip/hip_runtime.h>

typedef float v4f __attribute__((ext_vector_type(4)));
typedef float v2f __attribute__((ext_vector_type(2)));
typedef float v8f __attribute__((ext_vector_type(8)));

// Problem geometry (from setup_inputs)
constexpr int Hh   = 32;     // heads
constexpr int BUF  = 8192;   // ring buffer slots (2^13)
constexpr int Sn   = 1024;   // new tokens
constexpr int Dd   = 128;    // head dim (32 float4 per row)
constexpr int Wndw = BUF / 2;

// ---------------------------------------------------------------------------
// Fused ring-buffer scatter + copy.
// One wave32 = one (head, slot) row of 128 floats = 32 x b128 per wave.
// Slot s holds new token i = (s - start_pos) mod BUF if i < S, else old cache
// data. The select is wave-uniform, so no divergence; every output byte is
// written exactly once (no write-then-overwrite double traffic).
// ---------------------------------------------------------------------------
__global__ __launch_bounds__(256) void kv_copy_scatter(
    const float* __restrict__ cache, const float* __restrict__ val,
    const int* __restrict__ start_pos, float* __restrict__ out)
{
    const int sp   = start_pos[0];
    const int lane = threadIdx.x & 31;
    const int wave = threadIdx.x >> 5;
    const int row  = blockIdx.x * 8 + wave;        // [0, H*BUF)
    const int h    = row >> 13;                    // row / BUF
    const int s    = row & (BUF - 1);              // row % BUF

    int i = s - (sp & (BUF - 1));                  // token index landing on slot s
    if (i < 0) i += BUF;
    const bool written = (i < Sn);

    const float* src = written ? (val   + ((size_t)h * Sn  + i) * Dd)
                               : (cache + ((size_t)h * BUF + s) * Dd);
    float* dst = out + ((size_t)h * BUF + s) * Dd;

    // Streaming data: bypass L2 retention (working set 544MB >> 192MB L2).
    v4f x = __builtin_nontemporal_load((const v4f*)src + lane);
    __builtin_nontemporal_store(x, (v4f*)dst + lane);
}

// ---------------------------------------------------------------------------
// Sliding-window mask via WMMA outer-difference.
// delta[q,s] = pos_q[q] - cache_pos[s] is rank-2:  A(16x4) @ B(4x16) with
//   A[m][:] = [pos_q[m], 1, 0, 0],  B[:][n] = [1, -cache_pos[n], 0, 0]
// computed exactly in f32 by V_WMMA_F32_16X16X4_F32 (one 16x16 tile / wave).
// A layout (ISA 7.12.2, 32-bit 16x4): lanes 0-15 -> M=lane, v0=K0, v1=K1;
// lanes 16-31 -> K2,K3 (zeroed here). B mirrors with rows striped over lanes.
// EXEC is all-ones: full 256-thread blocks, uniform control flow only.
// ---------------------------------------------------------------------------
__global__ __launch_bounds__(256) void mask_wmma(
    const int* __restrict__ start_pos, float* __restrict__ mask)
{
    const int sp      = start_pos[0];
    const int end_pos = sp + Sn;
    const int lws     = (end_pos - 1) & (BUF - 1);   // last_write_slot
    const int base    = end_pos - 1 - lws;           // current_cycle_base

    const int lane   = threadIdx.x & 31;
    const int wave   = threadIdx.x >> 5;
    const int tile   = blockIdx.x * 8 + wave;        // [0, (S/16)*(BUF/16))
    const int tile_s = tile & (BUF / 16 - 1);        // 512 column tiles
    const int tile_q = tile >> 9;
    const int q0     = tile_q * 16;
    const int s0     = tile_s * 16;

    const int  n  = lane & 15;                       // column within tile
    const int  sc = s0 + n;                          // absolute slot
    const int  cp = (sc <= lws) ? (base + sc) : (base - BUF + sc); // cache_pos
    const bool lo = lane < 16;

    v2f a, b;
    a.x = lo ? (float)(sp + q0 + lane) : 0.0f;       // K=0: pos_q[m]
    a.y = lo ? 1.0f : 0.0f;                          // K=1: 1
    b.x = lo ? 1.0f : 0.0f;                          // K=0 row: ones
    b.y = lo ? -(float)cp : 0.0f;                    // K=1 row: -cache_pos[n]
    v8f c = {};
    v8f d = __builtin_amdgcn_wmma_f32_16x16x4_f32(
        /*neg_a=*/false, a, /*neg_b=*/false, b,
        /*c_mod=*/(short)0, c, /*reuse_a=*/false, /*reuse_b=*/false);

    const float NEG_INF = -__builtin_inff();
    const bool  cp_ok   = (cp >= 0);
    // D layout: VGPR r, lanes 0-15 -> row M=r; lanes 16-31 -> row M=r+8.
    float* outp = mask + (size_t)(q0 + (lo ? 0 : 8)) * BUF + sc;
#pragma unroll
    for (int r = 0; r < 8; ++r) {
        const float delta  = d[r];
        const bool  attend = cp_ok && (delta >= 0.0f) && (delta < (float)Wndw);
        __builtin_nontemporal_store(attend ? 0.0f : NEG_INF,
                                    outp + (size_t)r * BUF);
    }
}

extern "C" void kernel_launch(void* const* d_in, const int* in_sizes, int n_in,
                              void* d_out, int out_size, void* d_ws, size_t ws_size,
                              hipStream_t stream) {
    const float* k_cache = (const float*)d_in[0];
    const float* v_cache = (const float*)d_in[1];
    const float* k_val   = (const float*)d_in[2];
    const float* v_val   = (const float*)d_in[3];
    const int*   sp      = (const int*)d_in[4];

    float* out   = (float*)d_out;
    float* k_out = out;
    float* v_out = out + (size_t)Hh * BUF * Dd;
    float* mask  = out + 2 * (size_t)Hh * BUF * Dd;

    // H*BUF rows per tensor, 8 rows (waves) per 256-thread block.
    const int copy_blocks = (Hh * BUF) / 8;          // 32768
    kv_copy_scatter<<<copy_blocks, 256, 0, stream>>>(k_cache, k_val, sp, k_out);
    kv_copy_scatter<<<copy_blocks, 256, 0, stream>>>(v_cache, v_val, sp, v_out);

    // (S/16)*(BUF/16) = 32768 tiles, 8 tiles (waves) per block.
    const int mask_blocks = (Sn / 16) * (BUF / 16) / 8;  // 4096
    mask_wmma<<<mask_blocks, 256, 0, stream>>>(sp, mask);
}